// RandomCenteredRotationWithBoundingBox_5016521802369
// MI455X (gfx1250) — compile-verified
//
#include <hip/hip_runtime.h>
#include <math.h>

// ---------------------------------------------------------------------------
// RandomCenteredRotationWithBoundingBox — MI455X (gfx1250)
//
// B=128, H=W=128, floormap 4ch, acumap 16ch, mask 1ch.
// Bandwidth-bound warp/gather (~352 MB traffic -> ~15us at 23.3 TB/s).
// Kernel 1: per-batch bbox reduction (plane staged to LDS via gfx1250 async
//           load-to-LDS, ASYNCcnt-tracked) + transform params + point moves.
// Kernel 2: per-pixel inverse warp; 20 bilinear channels share one set of
//           weights/addresses; NT stores keep inputs resident in L2.
// ---------------------------------------------------------------------------

#define GRID_N   128
#define PIX_M    0.15625f           // 20.0 / 128, exact in binary
#define HALF_N   64.0f
#define TWO_PI_F 6.28318530717958647692f
#define PI_2_F   1.57079632679489661923f

#if defined(__has_builtin)
#  if __has_builtin(__builtin_amdgcn_global_load_async_to_lds_b128)
#    define HAVE_ASYNC_LDS 1
#  endif
#  if __has_builtin(__builtin_amdgcn_s_wait_asynccnt)
#    define HAVE_WAIT_ASYNC_BUILTIN 1
#  endif
#endif

typedef int v4i __attribute__((ext_vector_type(4)));

struct Params { float c, s, tx, ty; };

__global__ __launch_bounds__(256) void bbox_params_kernel(
    const float* __restrict__ floormap,
    const float* __restrict__ src,
    const float* __restrict__ rec,
    const float* __restrict__ u,
    const int*   __restrict__ rot_choice,
    Params*      __restrict__ params,
    float*       __restrict__ out_src,
    float*       __restrict__ out_rec)
{
    __shared__ int sxmin, sxmax, symin, symax;
    const int b   = blockIdx.x;
    const int tid = threadIdx.x;

    if (tid == 0) { sxmin = GRID_N; sxmax = -1; symin = GRID_N; symax = -1; }

    // channel BBOX_CH = 1 of floormap[b]
    const float* plane = floormap + ((size_t)(b * 4 + 1) << 14);

#ifdef HAVE_ASYNC_LDS
    // Stage the 64KB plane into LDS with gfx1250 async copies (ASYNCcnt).
    __shared__ float tile[GRID_N * GRID_N];
    {
        typedef __attribute__((address_space(1))) v4i g_v4i;
        typedef __attribute__((address_space(3))) v4i l_v4i;
        g_v4i* gp = (g_v4i*)plane;
        l_v4i* lp = (l_v4i*)tile;
        #pragma unroll
        for (int i = 0; i < 16; ++i) {
            const int e4 = (i << 8) | tid;        // 16-byte chunk index
            __builtin_amdgcn_global_load_async_to_lds_b128(
                gp + e4, lp + e4, /*offset=*/0, /*cpol=*/0);
        }
    }
#ifdef HAVE_WAIT_ASYNC_BUILTIN
    __builtin_amdgcn_s_wait_asynccnt(0);
#else
    asm volatile("s_wait_asynccnt 0" ::: "memory");
#endif
    __syncthreads();
    const float* scan_src = tile;
#else
    __syncthreads();
    const float* scan_src = plane;
#endif

    int xmin_t = GRID_N, xmax_t = -1, ymin_t = GRID_N, ymax_t = -1;
    #pragma unroll 4
    for (int i = 0; i < 64; ++i) {
        const int idx = tid + (i << 8);           // coalesced, stride 256
        const float v = scan_src[idx];
        if (v != 0.0f) {
            const int x = idx & 127;
            const int y = idx >> 7;
            xmin_t = min(xmin_t, x); xmax_t = max(xmax_t, x);
            ymin_t = min(ymin_t, y); ymax_t = max(ymax_t, y);
        }
    }
    atomicMin(&sxmin, xmin_t); atomicMax(&sxmax, xmax_t);
    atomicMin(&symin, ymin_t); atomicMax(&symax, ymax_t);
    __syncthreads();

    if (tid == 0) {
        int xmin = sxmin, xmax = sxmax, ymin = symin, ymax = symax;
        // argmax-of-all-false semantics in the reference -> full range
        if (xmax < 0) { xmin = 0; xmax = GRID_N - 1; }
        if (ymax < 0) { ymin = 0; ymax = GRID_N - 1; }

        const float xminf = (float)xmin, xmaxf = (float)xmax;
        const float yminf = (float)ymin, ymaxf = (float)ymax;

        const float hw   = ceilf((xmaxf - xminf) * 0.5f);
        const float hh   = ceilf((ymaxf - yminf) * 0.5f);
        const float hipo = sqrtf(hw * hw + hh * hh);

        const float angle = (hipo < HALF_N)
                              ? (u[b] * TWO_PI_F)
                              : ((float)rot_choice[b] * PI_2_F);

        const float c = cosf(angle);
        const float s = sinf(angle);
        const float tx = floorf(HALF_N - (xminf + xmaxf) * 0.5f);
        const float ty = floorf(HALF_N - (yminf + ymaxf) * 0.5f);

        Params p; p.c = c; p.s = s; p.tx = tx; p.ty = ty;
        params[b] = p;

        const float tmx = tx * PIX_M;
        const float tmy = ty * PIX_M;

        {   // src point
            const float px = src[b * 3 + 0] + tmy;
            const float py = src[b * 3 + 1] - tmx;
            out_src[b * 3 + 0] = c * px - s * py;
            out_src[b * 3 + 1] = s * px + c * py;
            out_src[b * 3 + 2] = src[b * 3 + 2];
        }
        {   // rec point
            const float px = rec[b * 3 + 0] + tmy;
            const float py = rec[b * 3 + 1] - tmx;
            out_rec[b * 3 + 0] = c * px - s * py;
            out_rec[b * 3 + 1] = s * px + c * py;
            out_rec[b * 3 + 2] = rec[b * 3 + 2];
        }
    }
}

__global__ __launch_bounds__(256) void warp_kernel(
    const float*  __restrict__ fm_in,
    const float*  __restrict__ am_in,
    const float*  __restrict__ mk_in,
    const Params* __restrict__ params,
    float*        __restrict__ fm_out,
    float*        __restrict__ am_out,
    float*        __restrict__ mk_out)
{
    // 64 blocks of 256 pixels per batch -> b is block-uniform (scalar loads)
    const int b   = blockIdx.x >> 6;
    const int pix = ((blockIdx.x & 63) << 8) | threadIdx.x;
    const int x   = pix & 127;
    const int y   = pix >> 7;

    const Params p = params[b];

    const float dx = (float)x - 63.5f;
    const float dy = (float)y - 63.5f;
    const float qx =  p.c * dx + p.s * dy + 63.5f - p.tx;
    const float qy = -p.s * dx + p.c * dy + 63.5f - p.ty;

    // ---- bilinear setup (shared by 20 channels) ----
    const float x0f = floorf(qx);
    const float y0f = floorf(qy);
    const float wx  = qx - x0f;
    const float wy  = qy - y0f;
    const int x0i = (int)x0f;
    const int y0i = (int)y0f;
    const bool bvalid = (x0i >= 0) && (x0i <= GRID_N - 2) &&
                        (y0i >= 0) && (y0i <= GRID_N - 2);
    const int x0c = min(max(x0i,     0), GRID_N - 1);
    const int x1c = min(max(x0i + 1, 0), GRID_N - 1);
    const int y0c = min(max(y0i,     0), GRID_N - 1);
    const int y1c = min(max(y0i + 1, 0), GRID_N - 1);
    const int i00 = (y0c << 7) | x0c;
    const int i01 = (y0c << 7) | x1c;
    const int i10 = (y1c << 7) | x0c;
    const int i11 = (y1c << 7) | x1c;
    const float wx1 = 1.0f - wx;
    const float wy1 = 1.0f - wy;

    // Warm caches for the big tensor's gather lines while we do the
    // floormap channels (gfx1250 global_prefetch_b8).
    const float* ap = am_in + ((size_t)(b << 4) << 14);
    __builtin_prefetch(ap + i00, 0, 3);
    __builtin_prefetch(ap + i01, 0, 3);
    __builtin_prefetch(ap + i10, 0, 3);
    __builtin_prefetch(ap + i11, 0, 3);

    // ---- floormap: 4 channels, bilinear, fill 0 ----
    const float* fp = fm_in  + ((size_t)(b << 2) << 14);
    float*       fo = fm_out + ((size_t)(b << 2) << 14) + pix;
    #pragma unroll
    for (int ch = 0; ch < 4; ++ch) {
        const float v00 = fp[i00], v01 = fp[i01];
        const float v10 = fp[i10], v11 = fp[i11];
        const float top = wx1 * v00 + wx * v01;
        const float bot = wx1 * v10 + wx * v11;
        const float val = wy1 * top + wy * bot;
        __builtin_nontemporal_store(bvalid ? val : 0.0f, fo);
        fp += GRID_N * GRID_N;
        fo += GRID_N * GRID_N;
    }

    // ---- acumap: 16 channels, bilinear, fill NaN ----
    const float NANF = __builtin_nanf("");
    float* ao = am_out + ((size_t)(b << 4) << 14) + pix;
    #pragma unroll
    for (int ch = 0; ch < 16; ++ch) {
        const float v00 = ap[i00], v01 = ap[i01];
        const float v10 = ap[i10], v11 = ap[i11];
        const float top = wx1 * v00 + wx * v01;
        const float bot = wx1 * v10 + wx * v11;
        const float val = wy1 * top + wy * bot;
        __builtin_nontemporal_store(bvalid ? val : NANF, ao);
        ap += GRID_N * GRID_N;
        ao += GRID_N * GRID_N;
    }

    // ---- mask: 1 channel, nearest (round half-to-even == rintf), fill 0 ----
    const float xr = rintf(qx);
    const float yr = rintf(qy);
    const bool nvalid = (xr >= 0.0f) && (xr <= (float)(GRID_N - 1)) &&
                        (yr >= 0.0f) && (yr <= (float)(GRID_N - 1));
    const int xn = min(max((int)xr, 0), GRID_N - 1);
    const int yn = min(max((int)yr, 0), GRID_N - 1);
    const float mv = mk_in[((size_t)b << 14) + (yn << 7) + xn];
    __builtin_nontemporal_store(nvalid ? mv : 0.0f,
                                mk_out + ((size_t)b << 14) + pix);
}

extern "C" void kernel_launch(void* const* d_in, const int* in_sizes, int n_in,
                              void* d_out, int out_size, void* d_ws, size_t ws_size,
                              hipStream_t stream)
{
    // setup_inputs() order: src, rec, floormap, acumap, mask, u, rot_choice
    const float* src = (const float*)d_in[0];
    const float* rec = (const float*)d_in[1];
    const float* fm  = (const float*)d_in[2];
    const float* am  = (const float*)d_in[3];
    const float* mk  = (const float*)d_in[4];
    const float* u   = (const float*)d_in[5];
    const int*   rc  = (const int*)  d_in[6];

    const int B = in_sizes[0] / 3;   // 128

    // Outputs concatenated flat in return order:
    // move(src)[B*3], move(rec)[B*3], fm[B*4*HW], am[B*16*HW], mk[B*1*HW]
    float* out      = (float*)d_out;
    float* out_src  = out;
    float* out_rec  = out + (size_t)B * 3;
    float* out_fm   = out + (size_t)B * 6;
    float* out_am   = out_fm + (size_t)B * 4  * GRID_N * GRID_N;
    float* out_mk   = out_am + (size_t)B * 16 * GRID_N * GRID_N;

    Params* params = (Params*)d_ws;   // B * 16 bytes

    bbox_params_kernel<<<B, 256, 0, stream>>>(fm, src, rec, u, rc,
                                              params, out_src, out_rec);
    warp_kernel<<<B * 64, 256, 0, stream>>>(fm, am, mk, params,
                                            out_fm, out_am, out_mk);
}